// Attention_38809324487275
// MI455X (gfx1250) — compile-verified
//
#include <hip/hip_runtime.h>
#include <hip/hip_bf16.h>
#include <math.h>

// Decode attention for MI455X (gfx1250).
// Memory-bound problem (~512MB traffic, ~200 MFLOP): all GEMVs are streamed
// with coalesced f32 loads; per-head logits tiles use V_WMMA_F32_16X16X4_F32
// with K tiles staged into LDS by the Tensor Data Mover (tensor_load_to_lds),
// double-buffered on TENSORcnt.

#define D_MODEL 4096
#define NHEADS  32
#define HDIM    128
#define SEQ     4096
#define KMASK   (-2.3819763e+38f)
#define KMASK_HALF (-1.19098815e+38f)

typedef float v2f __attribute__((ext_vector_type(2)));
typedef float v8f __attribute__((ext_vector_type(8)));
typedef unsigned int u32x4 __attribute__((ext_vector_type(4)));
typedef int i32x4 __attribute__((ext_vector_type(4)));
typedef int i32x8 __attribute__((ext_vector_type(8)));

// Workspace layout (float offsets). Total 90112 floats = 352 KB.
#define WS_QKV   0        // 3*32*128 = 12288  raw qkv accumulators
#define WS_QROT  12288    // 32*128   = 4096   roped+scaled q
#define WS_PART  16384    // 32*16*136 = 69632 flash partials {m,l,pad2,acc[128]}
#define WS_ENC   86016    // 32*128   = 4096   combined encoded

// ---------------------------------------------------------------- zero scratch
__global__ void k_zero(float* __restrict__ qkv, float* __restrict__ aout) {
  int i = blockIdx.x * blockDim.x + threadIdx.x;
  if (i < 3 * NHEADS * HDIM) qkv[i] = 0.f;
  if (i < D_MODEL)           aout[i] = 0.f;
}

// ------------------------------------------------- QKV projection (GEMV, M=1)
__global__ void k_qkv_proj(const float* __restrict__ x,
                           const float* __restrict__ w_qkv,
                           float* __restrict__ qkv) {
  const int GSPLIT = 8;
  int sn   = blockIdx.x / GSPLIT;   // 0..95  (= s*32+n)
  int part = blockIdx.x % GSPLIT;   // 0..7
  int h    = threadIdx.x;           // 0..127
  const float* wp = w_qkv + ((size_t)sn * D_MODEL + (size_t)part * 512) * HDIM + h;
  const float* xp = x + part * 512;
  float acc = 0.f;
#pragma unroll 8
  for (int d = 0; d < 512; ++d) {
    if ((d & 15) == 0)
      __builtin_prefetch(wp + (size_t)64 * HDIM, 0, 0);  // global_prefetch_b8
    acc = fmaf(xp[d], wp[0], acc);
    wp += HDIM;
  }
  atomicAdd(&qkv[sn * HDIM + h], acc);
}

// ----------------------------- RoPE(q,k), scale q, scatter k/v row into caches
__global__ void k_rope_scatter(const float* __restrict__ qkv,
                               const int* __restrict__ segpos,
                               const int* __restrict__ tstep,
                               float* __restrict__ q_rot,
                               float* __restrict__ kout,
                               float* __restrict__ vout) {
  int n = blockIdx.x;      // head
  int i = threadIdx.x;     // 0..63 (rotation pair index)
  float pos = (float)segpos[0];
  int   ts  = tstep[0];
  const float* q = qkv + (0 * NHEADS + n) * HDIM;
  const float* k = qkv + (1 * NHEADS + n) * HDIM;
  const float* v = qkv + (2 * NHEADS + n) * HDIM;
  float timescale = __powf(10000.f, (float)i * (1.f / 64.f));
  float sv, cv;
  __sincosf(pos / timescale, &sv, &cv);
  const float qscale = 0.08838834764831845f;  // 128^-0.5
  float q1 = q[i], q2 = q[i + 64];
  float k1 = k[i], k2 = k[i + 64];
  q_rot[n * HDIM + i]      = (q1 * cv - q2 * sv) * qscale;
  q_rot[n * HDIM + i + 64] = (q2 * cv + q1 * sv) * qscale;
  size_t row = ((size_t)ts * NHEADS + n) * HDIM;
  kout[row + i]      = k1 * cv - k2 * sv;
  kout[row + i + 64] = k2 * cv + k1 * sv;
  vout[row + i]      = v[i];
  vout[row + i + 64] = v[i + 64];
}

// ---------------------------------------------------- TDM K-tile load (16x128)
// D# per CDNA5 ISA:
//   group0: count=1, lds_addr, global_addr[56:0], type=2
//   group1: data_size=4B, pad_enable=1 (interval 128 DWORDs -> +4 DWORDs, i.e.
//           LDS row stride 132 floats), tensor_dim0=128, tensor_dim1=16,
//           tile_dim0=128, tile_dim1=16, tensor_dim0_stride=4096 elements.
__device__ __forceinline__ void tdm_load_k_tile(unsigned lds_addr,
                                                const float* gptr) {
  unsigned long long ga = (unsigned long long)(uintptr_t)gptr;
  u32x4 g0 = { 1u,                                   // count=1, user mode
               lds_addr,                             // LDS byte address
               (unsigned)ga,                         // global_addr[31:0]
               ((unsigned)(ga >> 32) & 0x01FFFFFFu)  // global_addr[56:32]
                   | 0x80000000u };                  // type=2 (bits 127:126=10b)
  // dword0: data_size=2(4B)<<16 | pad_enable<<20 | pad_interval=6<<22 |
  //         pad_amount=3<<25  (=> 0x07920000)
  i32x8 g1 = { (int)0x07920000,
               128 << 16,        // tensor_dim0 = 128   (bits 79:48, low half)
               16 << 16,         // tensor_dim1 = 16    (bits 111:80, low half)
               128 << 16,        // tile_dim0  = 128    (bits 127:112)
               16,               // tile_dim1  = 16     (bits 143:128)
               4096,             // tensor_dim0_stride  (bits 207:160, low 32)
               0, 0 };
  i32x4 z4 = { 0, 0, 0, 0 };          // groups 2/3 unused (2D tensor)
  i32x8 z8 = { 0, 0, 0, 0, 0, 0, 0, 0 };
  __builtin_amdgcn_tensor_load_to_lds(g0, g1, z4, z4, z8, 0);
}

// ------------------------------- flash-decode partials, logits via f32 WMMA
// One wave (32 threads, EXEC all ones) per (head, 256-row chunk).
// K tiles arrive via TDM (double buffered, TENSORcnt); logits via 32
// v_wmma_f32_16x16x4_f32 per tile; P*V via coalesced float4 FMAs.
__global__ __launch_bounds__(32) void k_attn_partial(
    const float* __restrict__ q_rot, const float* __restrict__ kout,
    const float* __restrict__ vout, const float* __restrict__ mask,
    float* __restrict__ part) {
  __shared__ float q_s[HDIM];
  __shared__ float lg[16];
  __shared__ float kt2[2 * 16 * 132];  // two 16x128 tiles, row stride 132

  int head  = blockIdx.x >> 4;
  int chunk = blockIdx.x & 15;
  int s0    = chunk * 256;
  int lane  = threadIdx.x;

  ((float4*)q_s)[lane] = ((const float4*)(q_rot + head * HDIM))[lane];

  const float* kbase = kout + (size_t)head * HDIM;  // + s*4096 per row
  unsigned lds_buf[2];
  lds_buf[0] = (unsigned)(uintptr_t)(&kt2[0]);
  lds_buf[1] = (unsigned)(uintptr_t)(&kt2[16 * 132]);

  // prologue: TDM-load tile 0
  tdm_load_k_tile(lds_buf[0], kbase + (size_t)s0 * 4096);

  float  m = -3.0e38f;
  float  l = 0.f;
  float4 acc = make_float4(0.f, 0.f, 0.f, 0.f);

  // WMMA A-operand layout: lanes 0-15 rows M0..15 {K,K+1}, lanes 16-31 {K+2,K+3}
  int row = lane & 15;
  int ko  = (lane >> 4) << 1;

  for (int tile = 0; tile < 16; ++tile) {
    int sb = s0 + tile * 16;
    if (tile < 15) {
      // previous reads of the buffer we are about to overwrite must be done
      asm volatile("s_wait_dscnt 0x0" ::: "memory");
      tdm_load_k_tile(lds_buf[(tile + 1) & 1],
                      kbase + (size_t)(sb + 16) * 4096);
      __builtin_amdgcn_s_wait_tensorcnt(1);  // current tile complete
    } else {
      __builtin_amdgcn_s_wait_tensorcnt(0);
    }
    __syncthreads();

    const float* kt = &kt2[(tile & 1) * (16 * 132)];

    v8f c = {0.f, 0.f, 0.f, 0.f, 0.f, 0.f, 0.f, 0.f};
#pragma unroll
    for (int kc = 0; kc < 32; ++kc) {
      int kk = kc * 4 + ko;
      v2f a, b;
      a[0] = kt[row * 132 + kk];
      a[1] = kt[row * 132 + kk + 1];
      b[0] = q_s[kk];          // q broadcast across all 16 B-columns
      b[1] = q_s[kk + 1];
      c = __builtin_amdgcn_wmma_f32_16x16x4_f32(
          false, a, false, b, (short)0, c, false, false);
    }
    // D: lanes 0-15 vgpr r -> row r, lanes 16-31 vgpr r -> row r+8
    if (lane == 0)  { for (int r = 0; r < 8; ++r) lg[r]     = c[r]; }
    if (lane == 16) { for (int r = 0; r < 8; ++r) lg[8 + r] = c[r]; }
    __syncthreads();

    // masked online softmax + P*V
    float lv[16];
    float nm = m;
#pragma unroll
    for (int r = 0; r < 16; ++r) {
      float lo = (mask[sb + r] >= KMASK_HALF) ? lg[r] : KMASK;
      lv[r] = lo;
      nm = fmaxf(nm, lo);
    }
    float scale = __expf(m - nm);
    l *= scale;
    acc.x *= scale; acc.y *= scale; acc.z *= scale; acc.w *= scale;
#pragma unroll
    for (int r = 0; r < 16; ++r) {
      float p = __expf(lv[r] - nm);
      l += p;
      const float4* vr =
          (const float4*)(vout + ((size_t)(sb + r) * NHEADS + head) * HDIM);
      float4 vv = vr[lane];
      acc.x = fmaf(p, vv.x, acc.x);
      acc.y = fmaf(p, vv.y, acc.y);
      acc.z = fmaf(p, vv.z, acc.z);
      acc.w = fmaf(p, vv.w, acc.w);
    }
    m = nm;
    __syncthreads();
  }

  float* pp = part + (size_t)(head * 16 + chunk) * 136;
  if (lane == 0) { pp[0] = m; pp[1] = l; }
  ((float4*)(pp + 4))[lane] = acc;   // pp+4 keeps 16B alignment
}

// ------------------------------------------- combine flash partials per head
__global__ __launch_bounds__(32) void k_combine(const float* __restrict__ part,
                                                float* __restrict__ enc) {
  int head = blockIdx.x;
  int lane = threadIdx.x;
  float M = -3.0e38f;
  for (int c = 0; c < 16; ++c)
    M = fmaxf(M, part[(size_t)(head * 16 + c) * 136]);
  float  L = 0.f;
  float4 e = make_float4(0.f, 0.f, 0.f, 0.f);
  for (int c = 0; c < 16; ++c) {
    const float* pp = part + (size_t)(head * 16 + c) * 136;
    float w = __expf(pp[0] - M);
    L += pp[1] * w;
    float4 a = ((const float4*)(pp + 4))[lane];
    e.x = fmaf(a.x, w, e.x);
    e.y = fmaf(a.y, w, e.y);
    e.z = fmaf(a.z, w, e.z);
    e.w = fmaf(a.w, w, e.w);
  }
  float inv = 1.f / L;
  e.x *= inv; e.y *= inv; e.z *= inv; e.w *= inv;
  ((float4*)(enc + head * HDIM))[lane] = e;
}

// --------------------------------- output projection: out[d] = enc . Wout[:,d]
__global__ void k_out_proj(const float* __restrict__ enc,
                           const float* __restrict__ w_out,
                           float* __restrict__ aout) {
  int dblk  = blockIdx.x >> 2;
  int ipart = blockIdx.x & 3;
  int d     = dblk * 128 + threadIdx.x;
  const float* wp = w_out + (size_t)(ipart * 1024) * D_MODEL + d;
  const float* ep = enc + ipart * 1024;
  float acc = 0.f;
#pragma unroll 8
  for (int i = 0; i < 1024; ++i) {
    acc = fmaf(ep[i], wp[0], acc);
    wp += D_MODEL;
  }
  atomicAdd(&aout[d], acc);
}

// ---------------------------------------------------------------------- launch
extern "C" void kernel_launch(void* const* d_in, const int* in_sizes, int n_in,
                              void* d_out, int out_size, void* d_ws, size_t ws_size,
                              hipStream_t stream) {
  const float* x      = (const float*)d_in[0];
  const float* w_qkv  = (const float*)d_in[1];
  const float* w_out  = (const float*)d_in[2];
  const float* kc     = (const float*)d_in[3];
  const float* vc     = (const float*)d_in[4];
  const float* mask   = (const float*)d_in[5];
  const int*   segpos = (const int*)d_in[6];
  const int*   tstep  = (const int*)d_in[7];

  float* kout = (float*)d_out;                       // [4096,32,128]
  float* vout = kout + (size_t)SEQ * NHEADS * HDIM;  // [4096,32,128]
  float* aout = vout + (size_t)SEQ * NHEADS * HDIM;  // [1,4096]

  float* ws   = (float*)d_ws;
  float* qkv  = ws + WS_QKV;
  float* qrot = ws + WS_QROT;
  float* part = ws + WS_PART;
  float* enc  = ws + WS_ENC;

  const size_t cache_bytes = (size_t)SEQ * NHEADS * HDIM * sizeof(float);

  k_zero<<<64, 256, 0, stream>>>(qkv, aout);
  k_qkv_proj<<<96 * 8, 128, 0, stream>>>(x, w_qkv, qkv);
  (void)hipMemcpyAsync(kout, kc, cache_bytes, hipMemcpyDeviceToDevice, stream);
  (void)hipMemcpyAsync(vout, vc, cache_bytes, hipMemcpyDeviceToDevice, stream);
  k_rope_scatter<<<NHEADS, 64, 0, stream>>>(qkv, segpos, tstep, qrot, kout, vout);
  k_attn_partial<<<NHEADS * 16, 32, 0, stream>>>(qrot, kout, vout, mask, part);
  k_combine<<<NHEADS, 32, 0, stream>>>(part, enc);
  k_out_proj<<<128, 128, 0, stream>>>(enc, w_out, aout);
}